// RetinaNetFocalLoss_16484084482810
// MI455X (gfx1250) — compile-verified
//
#include <hip/hip_runtime.h>
#include <hip/hip_bf16.h>

// ---------------- problem constants (match reference setup_inputs) ----------
#define B_BATCH 8
#define A_TOTAL 65536
#define T_TGT   64
#define C_CLS   80
#define ANCHORS_PER_BLOCK 64
#define BLOCKS_PER_BATCH (A_TOTAL / ANCHORS_PER_BLOCK)   // 1024
#define VEC4_PER_BLOCK   (ANCHORS_PER_BLOCK * C_CLS / 4) // 1280
#define THREADS 256

typedef float v2f __attribute__((ext_vector_type(2)));
typedef float v8f __attribute__((ext_vector_type(8)));

// Deterministic wave32 sum using V_WMMA_F32_16X16X4_F32 with B = ones.
// D[m][n] = sum_k A[m][k] (B all-ones makes the exact B layout irrelevant).
// With A=(v,0): rowsum_m = v[lane m] + v[lane m+16]; per-lane sum of the 8
// accumulator regs gives S_low (lanes 0-15) / S_high (lanes 16-31); xor-16 add
// yields the full 32-lane sum in every lane.
__device__ __forceinline__ float wave_reduce_wmma(float v) {
    v2f a; a.x = v;    a.y = 0.0f;
    v2f b; b.x = 1.0f; b.y = 1.0f;
    v8f c = {};
    v8f d = __builtin_amdgcn_wmma_f32_16x16x4_f32(
        /*neg_a=*/false, a, /*neg_b=*/false, b,
        /*c_mod=*/(short)0, c, /*reuse_a=*/false, /*reuse_b=*/false);
    float s = d[0] + d[1] + d[2] + d[3] + d[4] + d[5] + d[6] + d[7];
    s += __shfl_xor(s, 16, 32);
    return s;
}

__device__ __forceinline__ float huber1(float d) {
    float ad = fabsf(d);
    return (ad < 1.0f) ? 0.5f * d * d : ad - 0.5f;
}

// Focal element for logit x; pos = (this class is the matched target class).
// Uses hw-fast exp/log/rcp: argument of exp is -|x| <= 0 (no overflow) and
// 1+e in (1,2] keeps log well conditioned.
//   p  = sigmoid(x) ; r = 1/(1+e), e = exp(-|x|)
//   pt = pos ? 1-p : p  ==  (pos == (x>=0)) ? e*r : r
//   bce = softplus(x) - pos*x == log(1+e) + max(pos ? -x : x, 0)
__device__ __forceinline__ float focal_elem(float x, bool pos) {
    float e   = __expf(-fabsf(x));
    float ope = 1.0f + e;
    float r   = __builtin_amdgcn_rcpf(ope);
    float L   = __logf(ope);
    float pt  = (pos == (x >= 0.0f)) ? e * r : r;
    float w   = pt * pt * (pos ? 0.75f : 0.25f);   // gamma=2, alpha=0.25
    float bce = L + fmaxf(pos ? -x : x, 0.0f);
    return w * bce;
}

__global__ __launch_bounds__(THREADS)
void retina_main(const float* __restrict__ clas_preds,   // [B,A,C]
                 const float* __restrict__ bbox_preds,   // [B,A,4]
                 const int*   __restrict__ clas_tgts,    // [B,T] in [1,C]
                 const float* __restrict__ bbox_tgts,    // [B,T,4] tlbr
                 const float* __restrict__ anchors,      // [A,4] cthw
                 float* __restrict__ focal_part,         // [B*1024]
                 float* __restrict__ huber_part,         // [B*1024]
                 float* __restrict__ npos_part)          // [B*1024]
{
    const int b   = blockIdx.y;
    const int blk = blockIdx.x;
    const int tid = threadIdx.x;

    __shared__ float s_tlx[T_TGT], s_tly[T_TGT], s_brx[T_TGT], s_bry[T_TGT];
    __shared__ float s_cx[T_TGT], s_cy[T_TGT], s_w[T_TGT], s_h[T_TGT], s_area[T_TGT];
    __shared__ int   s_cls[T_TGT];
    __shared__ int   s_code[ANCHORS_PER_BLOCK]; // -1 ignore, 0 bg, 1..C match
    __shared__ float s_wf[8], s_wh[8];
    __shared__ int   s_wn[8];

    // ---- load + convert targets ----
    if (tid < T_TGT) {
        const float* bt = bbox_tgts + ((size_t)b * T_TGT + tid) * 4;
        float tlx = bt[0], tly = bt[1], brx = bt[2], bry = bt[3];
        s_tlx[tid] = tlx; s_tly[tid] = tly; s_brx[tid] = brx; s_bry[tid] = bry;
        float w = brx - tlx, h = bry - tly;
        s_cx[tid] = (tlx + brx) * 0.5f;
        s_cy[tid] = (tly + bry) * 0.5f;
        s_w[tid] = w; s_h[tid] = h; s_area[tid] = w * h;
        s_cls[tid] = clas_tgts[b * T_TGT + tid];
    }
    __syncthreads();

    // ---- matching + bbox smooth-L1 (one anchor per thread, threads 0..63) ----
    float huber_local = 0.0f;
    int   npos_local  = 0;
    if (tid < ANCHORS_PER_BLOCK) {
        const int a = blk * ANCHORS_PER_BLOCK + tid;
        const float acx = anchors[a * 4 + 0];
        const float acy = anchors[a * 4 + 1];
        const float aw  = anchors[a * 4 + 2];
        const float ah  = anchors[a * 4 + 3];
        const float a_tlx = acx - aw * 0.5f, a_tly = acy - ah * 0.5f;
        const float a_brx = acx + aw * 0.5f, a_bry = acy + ah * 0.5f;
        const float a_area = aw * ah;

        float best = -1.0f; int bi = 0;
        #pragma unroll 4
        for (int t = 0; t < T_TGT; ++t) {
            float ix = fminf(a_brx, s_brx[t]) - fmaxf(a_tlx, s_tlx[t]);
            float iy = fminf(a_bry, s_bry[t]) - fmaxf(a_tly, s_tly[t]);
            ix = fmaxf(ix, 0.0f); iy = fmaxf(iy, 0.0f);
            float inter = ix * iy;
            float uni   = a_area + s_area[t] - inter;
            float iou   = inter / (uni + 1e-8f);
            if (iou > best) { best = iou; bi = t; }   // first-max like argmax
        }

        int code;
        if (best > 0.5f) {
            code = s_cls[bi];                 // 1..C
            npos_local = 1;
            const float* bp = bbox_preds + ((size_t)b * A_TOTAL + a) * 4;
            // bbox_to_activ / BBOX_SCALE = [0.1,0.1,0.2,0.2]
            float tx = (s_cx[bi] - acx) / aw * 10.0f;
            float ty = (s_cy[bi] - acy) / ah * 10.0f;
            float tw = __builtin_logf(s_w[bi] / aw + 1e-8f) * 5.0f;
            float th = __builtin_logf(s_h[bi] / ah + 1e-8f) * 5.0f;
            huber_local = huber1(bp[0] - tx) + huber1(bp[1] - ty) +
                          huber1(bp[2] - tw) + huber1(bp[3] - th);
        } else if (best < 0.4f) {
            code = 0;                          // background
        } else {
            code = -1;                         // ignore
        }
        s_code[tid] = code;
    }
    __syncthreads();

    // ---- focal pass: 64 anchors x 80 classes = 1280 float4, coalesced ----
    float focal_local = 0.0f;
    const float4* cp = (const float4*)(clas_preds +
        ((size_t)b * A_TOTAL + (size_t)blk * ANCHORS_PER_BLOCK) * C_CLS);
    #pragma unroll
    for (int i = 0; i < VEC4_PER_BLOCK / THREADS; ++i) {
        int idx   = tid + i * THREADS;        // 0..1279
        int a_loc = idx / (C_CLS / 4);        // /20
        int c4    = idx % (C_CLS / 4);
        float4 x4 = cp[idx];
        int code  = s_code[a_loc];
        int ct    = code - 1;                 // -1 for bg -> never matches
        int cb    = c4 * 4;
        float v = focal_elem(x4.x, cb + 0 == ct)
                + focal_elem(x4.y, cb + 1 == ct)
                + focal_elem(x4.z, cb + 2 == ct)
                + focal_elem(x4.w, cb + 3 == ct);
        focal_local += (code >= 0) ? v : 0.0f;   // branchless ignore-mask
    }

    // ---- deterministic block reduction (WMMA wave sums + LDS combine) ----
    float wf = wave_reduce_wmma(focal_local);
    float wh = wave_reduce_wmma(huber_local);
    int   wn = (int)__popcll(__ballot(npos_local != 0));
    int wave = tid >> 5;
    if ((tid & 31) == 0) { s_wf[wave] = wf; s_wh[wave] = wh; s_wn[wave] = wn; }
    __syncthreads();
    if (tid == 0) {
        float f = 0.0f, h = 0.0f; int n = 0;
        #pragma unroll
        for (int w = 0; w < 8; ++w) { f += s_wf[w]; h += s_wh[w]; n += s_wn[w]; }
        size_t o = (size_t)b * BLOCKS_PER_BATCH + blk;
        focal_part[o] = f;
        huber_part[o] = h;
        npos_part[o]  = (float)n;
    }
}

__global__ __launch_bounds__(THREADS)
void retina_finish(const float* __restrict__ focal_part,
                   const float* __restrict__ huber_part,
                   const float* __restrict__ npos_part,
                   float* __restrict__ out)
{
    __shared__ float s_f[8], s_h[8], s_n[8];
    const int tid = threadIdx.x;
    float total = 0.0f;
    for (int b = 0; b < B_BATCH; ++b) {
        float f = 0.0f, h = 0.0f, n = 0.0f;
        for (int i = tid; i < BLOCKS_PER_BATCH; i += THREADS) {
            size_t o = (size_t)b * BLOCKS_PER_BATCH + i;
            f += focal_part[o]; h += huber_part[o]; n += npos_part[o];
        }
        f = wave_reduce_wmma(f);
        h = wave_reduce_wmma(h);
        n = wave_reduce_wmma(n);
        int wave = tid >> 5;
        if ((tid & 31) == 0) { s_f[wave] = f; s_h[wave] = h; s_n[wave] = n; }
        __syncthreads();
        if (tid == 0) {
            float F = 0.0f, H = 0.0f, N = 0.0f;
            #pragma unroll
            for (int w = 0; w < 8; ++w) { F += s_f[w]; H += s_h[w]; N += s_n[w]; }
            float bb = (N > 0.0f) ? H / fmaxf(N * 4.0f, 1.0f) : 0.0f;
            float fl = F / fmaxf(N, 1.0f);
            total += bb + fl;
        }
        __syncthreads();
    }
    if (tid == 0) out[0] = total / (float)B_BATCH;
}

extern "C" void kernel_launch(void* const* d_in, const int* in_sizes, int n_in,
                              void* d_out, int out_size, void* d_ws, size_t ws_size,
                              hipStream_t stream) {
    const float* clas_preds = (const float*)d_in[0];  // [B,A,C] f32
    const float* bbox_preds = (const float*)d_in[1];  // [B,A,4] f32
    const int*   clas_tgts  = (const int*)  d_in[2];  // [B,T] i32
    const float* bbox_tgts  = (const float*)d_in[3];  // [B,T,4] f32
    const float* anchors    = (const float*)d_in[4];  // [A,4] f32
    float* out = (float*)d_out;

    float* ws = (float*)d_ws;
    float* focal_part = ws;                                  // 8192 floats
    float* huber_part = ws + B_BATCH * BLOCKS_PER_BATCH;     // 8192 floats
    float* npos_part  = ws + 2 * B_BATCH * BLOCKS_PER_BATCH; // 8192 floats

    dim3 grid(BLOCKS_PER_BATCH, B_BATCH);
    retina_main<<<grid, THREADS, 0, stream>>>(
        clas_preds, bbox_preds, clas_tgts, bbox_tgts, anchors,
        focal_part, huber_part, npos_part);
    retina_finish<<<1, THREADS, 0, stream>>>(
        focal_part, huber_part, npos_part, out);
}